// SpatialAttention_47897475285072
// MI455X (gfx1250) — compile-verified
//
#include <hip/hip_runtime.h>
#include <stdint.h>

#define BATCH 128
#define NOBJ  1024
#define DIM   1024
#define SCALE 0.03125f            /* 1/sqrt(1024) */

#define CHUNKS 4
#define RPC    (NOBJ / CHUNKS)    /* 256 rows per chunk */
#define TILE_R 16                 /* rows per LDS tile  */
#define NTILES (RPC / TILE_R)     /* 16 tiles per chunk */
#define TPB    256
#define F4_PER_THREAD (TILE_R * (DIM / 4) / TPB)  /* 16 async b128 per thread per tile */

typedef __attribute__((ext_vector_type(2))) float v2f;
typedef __attribute__((ext_vector_type(8))) float v8f;
typedef int v4i __attribute__((vector_size(16)));  // matches async builtin's pointee

// ---------------------------------------------------------------------------
// CDNA5 fp32 WMMA: D(16x16) = A(16x4) * B(4x16) + C   (wave32)
// ---------------------------------------------------------------------------
__device__ __forceinline__ v8f wmma_f32x4(v2f a, v2f b, v8f c) {
#if __has_builtin(__builtin_amdgcn_wmma_f32_16x16x4_f32)
  return __builtin_amdgcn_wmma_f32_16x16x4_f32(
      /*neg_a=*/false, a, /*neg_b=*/false, b,
      /*c_mod=*/(short)0, c, /*reuse_a=*/false, /*reuse_b=*/false);
#else
  c[0] += a[0] * b[0] + a[1] * b[1];
  return c;
#endif
}

// ---------------------------------------------------------------------------
// Async global -> LDS copy (ASYNCcnt path), with sync fallback
// ---------------------------------------------------------------------------
__device__ __forceinline__ void async_load_f4(const float* g, float* l) {
#if __has_builtin(__builtin_amdgcn_global_load_async_to_lds_b128)
  __builtin_amdgcn_global_load_async_to_lds_b128(
      (__attribute__((address_space(1))) v4i*)(uintptr_t)g,
      (__attribute__((address_space(3))) v4i*)(uint32_t)(uintptr_t)l,
      0, 0);
#else
  *(float4*)l = *(const float4*)g;
#endif
}

template <int N>
__device__ __forceinline__ void wait_async() {
#if __has_builtin(__builtin_amdgcn_s_wait_asynccnt)
  __builtin_amdgcn_s_wait_asynccnt(N);
#else
  asm volatile("s_wait_asynccnt %0" ::"i"(N) : "memory");
#endif
}

// ---------------------------------------------------------------------------
// Kernel 1: qmix[b,d] = SCALE*(a1*(h0@q1_w^T + q1_b) + a2*(h1@q2_w^T + q2_b))
// One wave per 16x16 output tile; 512 tiles total.
// ---------------------------------------------------------------------------
__global__ __launch_bounds__(128) void qmix_kernel(
    const float* __restrict__ h, const float* __restrict__ q1_w,
    const float* __restrict__ q1_b, const float* __restrict__ q2_w,
    const float* __restrict__ q2_b, const float* __restrict__ alpha1,
    const float* __restrict__ alpha2, float* __restrict__ qmix) {
  const int wave = threadIdx.x >> 5;
  const int lane = threadIdx.x & 31;
  const int tile = blockIdx.x * 4 + wave;
  const int tm = tile >> 6;   // 8 M-tiles (B=128)
  const int tn = tile & 63;   // 64 N-tiles (D=1024)
  const int half = lane >> 4;
  const int l15 = lane & 15;
  const float a1 = alpha1[0] * SCALE;
  const float a2 = alpha2[0] * SCALE;

  const float* h0 = h;
  const float* h1 = h + BATCH * DIM;
  const int m = tm * 16 + l15;  // A row held by this lane
  const int n = tn * 16 + l15;  // B col held by this lane

  v8f c1 = {};
  v8f c2 = {};
  for (int k = 0; k < DIM; k += 4) {
    const int kb = k + 2 * half;  // K pair per half-wave (A layout: K0/K1 | K2/K3)
    v2f av1 = *(const v2f*)(h0 + (size_t)m * DIM + kb);
    v2f av2 = *(const v2f*)(h1 + (size_t)m * DIM + kb);
    v2f bv1 = *(const v2f*)(q1_w + (size_t)n * DIM + kb);  // B[k][n] = q1_w[n][k]
    v2f bv2 = *(const v2f*)(q2_w + (size_t)n * DIM + kb);
    c1 = wmma_f32x4(av1, bv1, c1);
    c2 = wmma_f32x4(av2, bv2, c2);
  }
  const float b1n = q1_b[n];
  const float b2n = q2_b[n];
#pragma unroll
  for (int i = 0; i < 8; ++i) {  // C layout: VGPR i -> M = i + 8*half, N = l15
    const int mr = tm * 16 + 8 * half + i;
    qmix[(size_t)mr * DIM + n] = a1 * (c1[i] + b1n) + a2 * (c2[i] + b2n);
  }
}

// ---------------------------------------------------------------------------
// Kernel 2: qeff = qmix @ wk_w   ( [128,1024] x [1024,1024], B not transposed )
// ---------------------------------------------------------------------------
__global__ __launch_bounds__(128) void qeff_kernel(const float* __restrict__ qmix,
                                                   const float* __restrict__ wk_w,
                                                   float* __restrict__ qeff) {
  const int wave = threadIdx.x >> 5;
  const int lane = threadIdx.x & 31;
  const int tile = blockIdx.x * 4 + wave;
  const int tm = tile >> 6;
  const int tn = tile & 63;
  const int half = lane >> 4;
  const int l15 = lane & 15;
  const int m = tm * 16 + l15;
  const int n = tn * 16 + l15;

  v8f c = {};
  for (int k = 0; k < DIM; k += 4) {
    const int kb = k + 2 * half;
    v2f av = *(const v2f*)(qmix + (size_t)m * DIM + kb);
    v2f bv;
    bv[0] = wk_w[(size_t)kb * DIM + n];        // B[k][n] = wk_w[k][n]
    bv[1] = wk_w[(size_t)(kb + 1) * DIM + n];
    c = wmma_f32x4(av, bv, c);
  }
#pragma unroll
  for (int i = 0; i < 8; ++i) {
    const int mr = tm * 16 + 8 * half + i;
    qeff[(size_t)mr * DIM + n] = c[i];
  }
}

// ---------------------------------------------------------------------------
// Kernel 3: streaming online-softmax over obj_embed (read once, async -> LDS,
// double-buffered). Block = (batch b, chunk c of 256 rows). All LDS traffic
// uses b128 accesses.
// ---------------------------------------------------------------------------
__global__ __launch_bounds__(TPB) void attn_stream_kernel(
    const float* __restrict__ obj, const float* __restrict__ qeff,
    float* __restrict__ pacc, float* __restrict__ pm, float* __restrict__ pl) {
  __shared__ __align__(16) float tile[2][TILE_R][DIM];  // 2 x 64 KB
  __shared__ float scores_s[TILE_R];
  __shared__ float w_s[TILE_R];

  const int b = blockIdx.x / CHUNKS;
  const int cch = blockIdx.x % CHUNKS;
  const int tid = threadIdx.x;
  const int wave = tid >> 5;
  const int lane = tid & 31;
  const size_t gchunk = ((size_t)b * NOBJ + (size_t)cch * RPC) * DIM;

  // each lane caches the 32 qeff values it needs for row dots (8 x float4)
  float4 qreg4[8];
#pragma unroll
  for (int i = 0; i < 8; ++i)
    qreg4[i] = *(const float4*)(qeff + (size_t)b * DIM + i * 128 + lane * 4);

  float4 acc = {0.f, 0.f, 0.f, 0.f};  // 4 consecutive columns per thread
  float run_m = -3.0e38f, run_l = 0.f;

  // prologue: issue tile 0
#pragma unroll
  for (int j = 0; j < TILE_R; ++j)
    async_load_f4(obj + gchunk + (size_t)j * DIM + tid * 4, &tile[0][j][tid * 4]);

  for (int t = 0; t < NTILES; ++t) {
    const int buf = t & 1;
    if (t + 1 < NTILES) {  // issue next tile into the other buffer
      const size_t gb = gchunk + (size_t)(t + 1) * TILE_R * DIM;
#pragma unroll
      for (int j = 0; j < TILE_R; ++j)
        async_load_f4(obj + gb + (size_t)j * DIM + tid * 4,
                      &tile[buf ^ 1][j][tid * 4]);
      wait_async<F4_PER_THREAD>();  // current tile's loads done (in-order)
    } else {
      wait_async<0>();
    }
    __syncthreads();

    // phase 1: row dot products (2 rows per wave, ds_load_b128)
#pragma unroll
    for (int rr = 0; rr < 2; ++rr) {
      const int r = wave * 2 + rr;
      float p = 0.f;
#pragma unroll
      for (int i = 0; i < 8; ++i) {
        const float4 tv = *(const float4*)&tile[buf][r][i * 128 + lane * 4];
        p += tv.x * qreg4[i].x + tv.y * qreg4[i].y + tv.z * qreg4[i].z +
             tv.w * qreg4[i].w;
      }
      p += __shfl_xor(p, 16);
      p += __shfl_xor(p, 8);
      p += __shfl_xor(p, 4);
      p += __shfl_xor(p, 2);
      p += __shfl_xor(p, 1);
      if (lane == 0) scores_s[r] = p;
    }
    __syncthreads();

    // phase 2: online softmax update (redundant per-thread, deterministic)
    const float sc = scores_s[lane & 15];
    float tmax = sc;
    tmax = fmaxf(tmax, __shfl_xor(tmax, 16));
    tmax = fmaxf(tmax, __shfl_xor(tmax, 8));
    tmax = fmaxf(tmax, __shfl_xor(tmax, 4));
    tmax = fmaxf(tmax, __shfl_xor(tmax, 2));
    tmax = fmaxf(tmax, __shfl_xor(tmax, 1));
    const float new_m = fmaxf(run_m, tmax);
    const float alpha = __expf(run_m - new_m);
    const float wv = __expf(sc - new_m);
    float tsum = wv;
    tsum += __shfl_xor(tsum, 16);
    tsum += __shfl_xor(tsum, 8);
    tsum += __shfl_xor(tsum, 4);
    tsum += __shfl_xor(tsum, 2);
    tsum += __shfl_xor(tsum, 1);
    tsum *= 0.5f;  // 16 distinct values duplicated across 32 lanes
    run_l = run_l * alpha + tsum;
    run_m = new_m;
    if (tid < TILE_R) w_s[tid] = wv;  // wave0 lanes 0..15 hold rows 0..15
    __syncthreads();

    // phase 3: rescale + weighted row accumulation (float4 per thread)
    acc.x *= alpha; acc.y *= alpha; acc.z *= alpha; acc.w *= alpha;
#pragma unroll
    for (int r = 0; r < TILE_R; ++r) {
      const float wr = w_s[r];
      const float4 tv = *(const float4*)&tile[buf][r][tid * 4];
      acc.x += wr * tv.x;
      acc.y += wr * tv.y;
      acc.z += wr * tv.z;
      acc.w += wr * tv.w;
    }
    __syncthreads();
  }

  const size_t pbase = ((size_t)b * CHUNKS + cch) * DIM;
  *(float4*)(pacc + pbase + tid * 4) = acc;
  if (tid == 0) {
    pm[b * CHUNKS + cch] = run_m;
    pl[b * CHUNKS + cch] = run_l;
  }
}

// ---------------------------------------------------------------------------
// Kernel 4: merge chunk partials -> s[b,d] = softmax-weighted sum of obj rows
// ---------------------------------------------------------------------------
__global__ __launch_bounds__(256) void combine_kernel(const float* __restrict__ pacc,
                                                      const float* __restrict__ pm,
                                                      const float* __restrict__ pl,
                                                      float* __restrict__ sdst) {
  const int b = blockIdx.x;
  const int tid = threadIdx.x;
  const float m0 = pm[b * 4 + 0], m1 = pm[b * 4 + 1];
  const float m2 = pm[b * 4 + 2], m3 = pm[b * 4 + 3];
  const float M = fmaxf(fmaxf(m0, m1), fmaxf(m2, m3));
  const float f0 = __expf(m0 - M), f1 = __expf(m1 - M);
  const float f2 = __expf(m2 - M), f3 = __expf(m3 - M);
  const float L = pl[b * 4 + 0] * f0 + pl[b * 4 + 1] * f1 +
                  pl[b * 4 + 2] * f2 + pl[b * 4 + 3] * f3;
  const float invL = 1.0f / L;

  const float4 v0 = *(const float4*)(pacc + ((size_t)b * 4 + 0) * DIM + tid * 4);
  const float4 v1 = *(const float4*)(pacc + ((size_t)b * 4 + 1) * DIM + tid * 4);
  const float4 v2 = *(const float4*)(pacc + ((size_t)b * 4 + 2) * DIM + tid * 4);
  const float4 v3 = *(const float4*)(pacc + ((size_t)b * 4 + 3) * DIM + tid * 4);
  float4 r;
  r.x = (v0.x * f0 + v1.x * f1 + v2.x * f2 + v3.x * f3) * invL;
  r.y = (v0.y * f0 + v1.y * f1 + v2.y * f2 + v3.y * f3) * invL;
  r.z = (v0.z * f0 + v1.z * f1 + v2.z * f2 + v3.z * f3) * invL;
  r.w = (v0.w * f0 + v1.w * f1 + v2.w * f2 + v3.w * f3) * invL;
  *(float4*)(sdst + (size_t)b * DIM + tid * 4) = r;
}

// ---------------------------------------------------------------------------
// Kernel 5: out = s @ wv_w^T + wv_b   (WMMA, same tiling as kernel 1)
// ---------------------------------------------------------------------------
__global__ __launch_bounds__(128) void out_kernel(const float* __restrict__ s,
                                                  const float* __restrict__ wv_w,
                                                  const float* __restrict__ wv_b,
                                                  float* __restrict__ out) {
  const int wave = threadIdx.x >> 5;
  const int lane = threadIdx.x & 31;
  const int tile = blockIdx.x * 4 + wave;
  const int tm = tile >> 6;
  const int tn = tile & 63;
  const int half = lane >> 4;
  const int l15 = lane & 15;
  const int m = tm * 16 + l15;
  const int n = tn * 16 + l15;

  v8f c = {};
  for (int k = 0; k < DIM; k += 4) {
    const int kb = k + 2 * half;
    v2f av = *(const v2f*)(s + (size_t)m * DIM + kb);
    v2f bv = *(const v2f*)(wv_w + (size_t)n * DIM + kb);  // B[k][n] = wv_w[n][k]
    c = wmma_f32x4(av, bv, c);
  }
  const float bn = wv_b[n];
#pragma unroll
  for (int i = 0; i < 8; ++i) {
    const int mr = tm * 16 + 8 * half + i;
    out[(size_t)mr * DIM + n] = c[i] + bn;  // [B,1,D] flat
  }
}

// ---------------------------------------------------------------------------
extern "C" void kernel_launch(void* const* d_in, const int* in_sizes, int n_in,
                              void* d_out, int out_size, void* d_ws, size_t ws_size,
                              hipStream_t stream) {
  (void)in_sizes; (void)n_in; (void)out_size; (void)ws_size;
  const float* obj    = (const float*)d_in[0];
  const float* h      = (const float*)d_in[1];
  const float* q1_w   = (const float*)d_in[2];
  const float* q1_b   = (const float*)d_in[3];
  const float* q2_w   = (const float*)d_in[4];
  const float* q2_b   = (const float*)d_in[5];
  const float* wk_w   = (const float*)d_in[6];
  /* d_in[7] = wk_b: drops out — softmax is shift-invariant per batch */
  const float* wv_w   = (const float*)d_in[8];
  const float* wv_b   = (const float*)d_in[9];
  const float* alpha1 = (const float*)d_in[10];
  const float* alpha2 = (const float*)d_in[11];
  float* out = (float*)d_out;

  float* ws   = (float*)d_ws;
  float* qmix = ws;                              // B*D
  float* qeff = qmix + BATCH * DIM;              // B*D
  float* pacc = qeff + BATCH * DIM;              // B*CHUNKS*D
  float* pm   = pacc + BATCH * CHUNKS * DIM;     // B*CHUNKS
  float* pl   = pm + BATCH * CHUNKS;             // B*CHUNKS
  float* sbuf = pl + BATCH * CHUNKS;             // B*D

  qmix_kernel<<<128, 128, 0, stream>>>(h, q1_w, q1_b, q2_w, q2_b, alpha1, alpha2, qmix);
  qeff_kernel<<<128, 128, 0, stream>>>(qmix, wk_w, qeff);
  attn_stream_kernel<<<BATCH * CHUNKS, TPB, 0, stream>>>(obj, qeff, pacc, pm, pl);
  combine_kernel<<<BATCH, 256, 0, stream>>>(pacc, pm, pl, sbuf);
  out_kernel<<<128, 128, 0, stream>>>(sbuf, wv_w, wv_b, out);
}